// MeanProbExtractor_yolov5_14353780703312
// MI455X (gfx1250) — compile-verified
//
#include <hip/hip_runtime.h>
#include <hip/hip_bf16.h>
#include <stdint.h>

#define YB      16
#define YN      25200
#define YNC     80
#define YSTRIDE 85
#define YCONF   0.25f
#define YIOU    0.45f
#define YMAXDET 300
#define YMAXWH  7680.0f
#define YTOPK   1024
#define YTILE   128
// NOTE: YB*YN == 403200 == 3150 * YTILE exactly -> all tiles are full, and
// YTILE*YSTRIDE == 10880 dwords is divisible by 4 (b128-friendly), with every
// tile base offset 16B-aligned (43,520 B per tile).

// ---- gfx1250 async global->LDS staging.
// b32 params (probe-confirmed): (as1 int*, as3 int*, imm, imm)
// b128 params (probe-confirmed): (as1 v4i*, as3 v4i*, imm, imm) where
//   v4i = int __attribute__((vector_size(16)))
#if defined(__HIP_DEVICE_COMPILE__) && \
    __has_builtin(__builtin_amdgcn_global_load_async_to_lds_b32) && \
    __has_builtin(__builtin_amdgcn_s_wait_asynccnt)
#define USE_ASYNC_LDS 1
typedef __attribute__((address_space(1))) int as1_int;
typedef __attribute__((address_space(3))) int as3_int;
typedef int yv4i __attribute__((__vector_size__(16)));
typedef __attribute__((address_space(1))) yv4i as1_v4i;
typedef __attribute__((address_space(3))) yv4i as3_v4i;
#if __has_builtin(__builtin_amdgcn_global_load_async_to_lds_b128)
#define HAVE_ASYNC_B128 1
#else
#define HAVE_ASYNC_B128 0
#endif
#else
#define USE_ASYNC_LDS 0
#endif

// Order-preserving float<->uint mapping (monotone for total order on floats)
__device__ __forceinline__ unsigned f2ord(float f) {
    unsigned u = __float_as_uint(f);
    return (u & 0x80000000u) ? ~u : (u | 0x80000000u);
}
__device__ __forceinline__ float ord2f(unsigned u) {
    unsigned v = (u & 0x80000000u) ? (u & 0x7FFFFFFFu) : ~u;
    return __uint_as_float(v);
}

// ---------------------------------------------------------------------------
// Kernel 1: per-anchor score/class.  One block = one 128-anchor tile.
// Bandwidth-bound streaming pass over the 137MB input; tile staged into LDS
// with gfx1250 async global->LDS copies (b128 wide variant when available:
// 512B per instruction vs 128B -> 4x lower issue pressure at 23.3 TB/s).
// ---------------------------------------------------------------------------
__global__ __launch_bounds__(256) void score_kernel(const float* __restrict__ in,
                                                    float* __restrict__ scores,
                                                    int* __restrict__ cls) {
    const int tileBase = blockIdx.x * YTILE;   // flat anchor index over B*N
    const int total    = YB * YN;
    const int t        = threadIdx.x;

#if USE_ASYNC_LDS
    __shared__ __align__(16) float buf[YTILE * YSTRIDE];   // 43,520 B
    const float* gsrc = in + (size_t)tileBase * YSTRIDE;
#if HAVE_ASYNC_B128
    for (int d = t * 4; d < YTILE * YSTRIDE; d += 256 * 4) {
        __builtin_amdgcn_global_load_async_to_lds_b128(
            (as1_v4i*)(gsrc + d), (as3_v4i*)(buf + d), 0, 0);
    }
#else
    for (int d = t; d < YTILE * YSTRIDE; d += 256) {
        __builtin_amdgcn_global_load_async_to_lds_b32(
            (as1_int*)(gsrc + d), (as3_int*)(buf + d), 0, 0);
    }
#endif
    __builtin_amdgcn_s_wait_asynccnt(0);
    __syncthreads();
    if (t < YTILE && tileBase + t < total) {
        const float* base = buf + t * YSTRIDE;
        float obj  = base[4];
        float best = base[5];
        int   bj   = 0;
        for (int c = 1; c < YNC; ++c) {
            float v = base[5 + c];
            if (v > best) { best = v; bj = c; }
        }
        float conf  = obj * best;
        bool  valid = (obj > YCONF) && (conf > YCONF);
        int a = tileBase + t;
        scores[a] = valid ? conf : -1.0f;
        cls[a]    = bj;
    }
#else
    if (t < YTILE) {
        int a = tileBase + t;
        if (a < total) {
            const float* base = in + (size_t)a * YSTRIDE;
            __builtin_prefetch(base + 32, 0, 0);   // pull tail cachelines early
            __builtin_prefetch(base + 64, 0, 0);
            float obj  = base[4];
            float best = base[5];
            int   bj   = 0;
            for (int c = 1; c < YNC; ++c) {
                float v = base[5 + c];
                if (v > best) { best = v; bj = c; }
            }
            float conf  = obj * best;
            bool  valid = (obj > YCONF) && (conf > YCONF);
            scores[a] = valid ? conf : -1.0f;
            cls[a]    = bj;
        }
    }
#endif
}

// ---------------------------------------------------------------------------
// Kernel 2: exact sorted top-1024 per batch.
// 4-pass radix-select (256-bin histograms in LDS) finds the exact 1024th
// largest key, then compaction + bitonic sort of 1024 packed 64-bit
// (score|~idx) keys in 8KB LDS.  Deterministic: keys are unique.
// ---------------------------------------------------------------------------
__global__ __launch_bounds__(256) void topk_kernel(const float* __restrict__ scores,
                                                   float* __restrict__ tsc,
                                                   int* __restrict__ tidx) {
    const int b = blockIdx.x;
    const int t = threadIdx.x;
    const float* s = scores + (size_t)b * YN;

    __shared__ unsigned hist[256];
    __shared__ unsigned s_prefix, s_pmask, s_want, s_cntG, s_cntE;
    __shared__ unsigned long long skey[YTOPK];   // 8 KB

    if (t == 0) { s_prefix = 0u; s_pmask = 0u; s_want = YTOPK; }

    for (int pass = 3; pass >= 0; --pass) {
        hist[t] = 0u;
        __syncthreads();
        const unsigned pm = s_pmask, pf = s_prefix;
        const int shift = pass * 8;
        for (int i = t; i < YN; i += 256) {
            if (i + 256 < YN) __builtin_prefetch(s + i + 256, 0, 0);
            unsigned u = f2ord(s[i]);
            if ((u & pm) == pf) atomicAdd(&hist[(u >> shift) & 0xFFu], 1u);
        }
        __syncthreads();
        if (t == 0) {
            unsigned rem = s_want;
            int v = 255;
            for (; v > 0; --v) { unsigned c = hist[v]; if (rem <= c) break; rem -= c; }
            s_prefix = pf | ((unsigned)v << shift);
            s_pmask  = pm | (0xFFu << shift);
            s_want   = rem;
        }
        __syncthreads();
    }

    const unsigned T    = s_prefix;         // exact key of the 1024th largest
    const unsigned want = s_want;           // how many ==T to take
    const unsigned G    = YTOPK - want;     // exact count of keys > T
    if (t == 0) { s_cntG = 0u; s_cntE = 0u; }
    __syncthreads();

    for (int i = t; i < YN; i += 256) {
        unsigned u = f2ord(s[i]);
        if (u > T) {
            unsigned p = atomicAdd(&s_cntG, 1u);
            if (p < YTOPK)
                skey[p] = ((unsigned long long)u << 32) | (unsigned)(0xFFFFFFFFu - (unsigned)i);
        } else if (u == T) {
            unsigned e = atomicAdd(&s_cntE, 1u);
            if (e < want)
                skey[G + e] = ((unsigned long long)u << 32) | (unsigned)(0xFFFFFFFFu - (unsigned)i);
        }
    }
    __syncthreads();

    // Bitonic sort, descending (ties -> lower original index first via ~idx).
    for (int k = 2; k <= YTOPK; k <<= 1) {
        for (int j = k >> 1; j > 0; j >>= 1) {
            for (int e = t; e < YTOPK; e += 256) {
                int ixj = e ^ j;
                if (ixj > e) {
                    unsigned long long a = skey[e], c = skey[ixj];
                    bool desc = ((e & k) == 0);
                    if (desc ? (a < c) : (a > c)) { skey[e] = c; skey[ixj] = a; }
                }
            }
            __syncthreads();
        }
    }

    for (int e = t; e < YTOPK; e += 256) {
        unsigned long long kk = skey[e];
        unsigned u = (unsigned)(kk >> 32);
        tsc[(size_t)b * YTOPK + e]  = ord2f(u);
        tidx[(size_t)b * YTOPK + e] = (int)(0xFFFFFFFFu - (unsigned)(kk & 0xFFFFFFFFu));
    }
}

// ---------------------------------------------------------------------------
// Kernel 3: per-batch NMS.  1024 threads; wave32 bitmask NMS:
//   row build  : thread t computes IoU(t, j) for all j, packs 32 words
//   greedy scan: one wave32 (32 lanes, one suppression word per lane)
//   compaction : ballot-based rank, first 300 kept -> det rows
// ---------------------------------------------------------------------------
__global__ __launch_bounds__(1024) void nms_kernel(const float* __restrict__ in,
                                                   const float* __restrict__ tsc,
                                                   const int* __restrict__ tidx,
                                                   const int* __restrict__ cls,
                                                   unsigned* __restrict__ maskG,
                                                   float* __restrict__ dets,
                                                   float* __restrict__ bstats) {
    const int b = blockIdx.x;
    const int t = threadIdx.x;

    __shared__ float4 boff[YTOPK];     // class-offset boxes (16 KB)
    __shared__ float4 borg[YTOPK];     // original boxes     (16 KB)
    __shared__ float  ssc[YTOPK];      // scores              (4 KB)
    __shared__ float  sclsf[YTOPK];    // class as float      (4 KB)
    __shared__ float  sterm[YTOPK];    // wh-terms            (4 KB)
    __shared__ unsigned suppInit[YTOPK / 32];
    __shared__ unsigned suppFin[YTOPK / 32];
    __shared__ unsigned waveCnt[YTOPK / 32];
    __shared__ unsigned s_n;
    __shared__ float    s_maxc;

    const float sc = tsc[(size_t)b * YTOPK + t];
    int idx = tidx[(size_t)b * YTOPK + t];
    if (idx < 0 || idx >= YN) idx = 0;
    const float* p = in + ((size_t)b * YN + idx) * YSTRIDE;
    const float cx = p[0], cy = p[1], w = p[2], h = p[3];
    const float x1 = cx - 0.5f * w, y1 = cy - 0.5f * h;
    const float x2 = cx + 0.5f * w, y2 = cy + 0.5f * h;
    const int   cj = cls[(size_t)b * YN + idx];
    const float off = (float)cj * YMAXWH;

    borg[t]  = make_float4(x1, y1, x2, y2);
    boff[t]  = make_float4(x1 + off, y1 + off, x2 + off, y2 + off);
    ssc[t]   = sc;
    sclsf[t] = (float)cj;

    unsigned long long bal = __ballot(sc <= 0.0f);   // pre-suppress invalid
    if ((t & 31) == 0) suppInit[t >> 5] = (unsigned)bal;
    if (t == 0) { s_n = 0u; s_maxc = 0.0f; }

    // zero this batch's det rows
    float* db = dets + (size_t)b * YMAXDET * 6;
    for (int i = t; i < YMAXDET * 6; i += 1024) db[i] = 0.0f;
    __syncthreads();

    // ---- IoU bitmask row for candidate t ----
    const float ax1 = x1 + off, ay1 = y1 + off, ax2 = x2 + off, ay2 = y2 + off;
    const float areaA = (x2 - x1) * (y2 - y1);
    unsigned* myrow = maskG + ((size_t)b * YTOPK + t) * 32;
    for (int wq = 0; wq < 32; ++wq) {
        unsigned bits = 0u;
        for (int k = 0; k < 32; ++k) {
            const int j = wq * 32 + k;              // uniform -> LDS broadcast
            float4 bb = boff[j];
            float xx1 = fmaxf(ax1, bb.x), yy1 = fmaxf(ay1, bb.y);
            float xx2 = fminf(ax2, bb.z), yy2 = fminf(ay2, bb.w);
            float iw = fmaxf(xx2 - xx1, 0.0f), ih = fmaxf(yy2 - yy1, 0.0f);
            float inter = iw * ih;
            float areaB = (bb.z - bb.x) * (bb.w - bb.y);
            float iou = inter / (areaA + areaB - inter + 1e-7f);
            if (iou > YIOU && j > t) bits |= (1u << k);
        }
        myrow[wq] = bits;
    }
    __threadfence_block();
    __syncthreads();

    // ---- greedy sequential scan: one wave, one supp word per lane ----
    if (t < 32) {
        unsigned suppw = suppInit[t];
        const unsigned* mbase = maskG + (size_t)b * YTOPK * 32;
        for (int i = 0; i < YTOPK; ++i) {
            unsigned wrd = __shfl(suppw, i >> 5, 32);
            bool active = ((wrd >> (i & 31)) & 1u) == 0u;
            if (active) {
                if (i + 2 < YTOPK) __builtin_prefetch(mbase + (size_t)(i + 2) * 32 + t, 0, 0);
                suppw |= mbase[(size_t)i * 32 + t];
            }
        }
        suppFin[t] = suppw;
    }
    __syncthreads();

    // ---- keep flags, ballot rank, compact to det rows ----
    const unsigned fin = suppFin[t >> 5];
    const bool keep = (ssc[t] > 0.0f) && (((fin >> (t & 31)) & 1u) == 0u);
    const int lane = t & 31, wv = t >> 5;
    unsigned wmask = (unsigned)__ballot(keep);
    if (lane == 0) waveCnt[wv] = (unsigned)__popc(wmask);
    __syncthreads();
    if (t == 0) {
        unsigned acc = 0u;
        for (int i = 0; i < YTOPK / 32; ++i) { unsigned c = waveCnt[i]; waveCnt[i] = acc; acc += c; }
        s_n = acc;
    }
    __syncthreads();
    const int rank = (int)waveCnt[wv] + __popc(wmask & ((1u << lane) - 1u));

    float term = 0.0f;
    if (keep && rank < YMAXDET) {
        float4 bo = borg[t];
        float  s  = ssc[t];
        float* r = db + rank * 6;
        r[0] = bo.x; r[1] = bo.y; r[2] = bo.z; r[3] = bo.w; r[4] = s; r[5] = sclsf[t];
        term = (fabsf(bo.x - bo.z) + fabsf(bo.y - bo.w)) * s;
        if (rank == 0) s_maxc = s;   // max kept score (sorted desc)
    }
    sterm[t] = term;
    __syncthreads();

    // deterministic tree reduction of wh-terms
    for (int stride = YTOPK / 2; stride > 0; stride >>= 1) {
        if (t < stride) sterm[t] += sterm[t + stride];
        __syncthreads();
    }
    if (t == 0) {
        unsigned n = s_n; if (n > YMAXDET) n = YMAXDET;
        float wh = (n > 0u) ? (sterm[0] / (2.0f * (float)(n > 0u ? n : 1u))) : 0.0f;
        bstats[b * 2 + 0] = s_maxc;
        bstats[b * 2 + 1] = wh;
    }
}

// ---------------------------------------------------------------------------
// Kernel 4: finalize -> out[0]=conf_addition, out[1]=w_and_h, out[2..]=det[2]
// ---------------------------------------------------------------------------
__global__ __launch_bounds__(256) void finalize_kernel(const float* __restrict__ dets,
                                                       const float* __restrict__ bstats,
                                                       float* __restrict__ out) {
    const int t = threadIdx.x;
    if (t == 0) {
        float cs = 0.0f, ws = 0.0f;
        for (int b = 0; b < YB; ++b) { cs += bstats[b * 2]; ws += bstats[b * 2 + 1]; }
        out[0] = cs / (float)YB;
        out[1] = ws / (float)YB;
    }
    const float* d2 = dets + (size_t)2 * YMAXDET * 6;
    for (int i = t; i < YMAXDET * 6; i += 256) out[2 + i] = d2[i];
}

// ---------------------------------------------------------------------------
extern "C" void kernel_launch(void* const* d_in, const int* in_sizes, int n_in,
                              void* d_out, int out_size, void* d_ws, size_t ws_size,
                              hipStream_t stream) {
    (void)in_sizes; (void)n_in; (void)out_size; (void)ws_size;
    const float* in  = (const float*)d_in[0];
    float*       out = (float*)d_out;

    char* ws = (char*)d_ws;
    size_t off = 0;
    float*    scores = (float*)(ws + off);    off += (size_t)YB * YN * sizeof(float);
    int*      cls    = (int*)(ws + off);      off += (size_t)YB * YN * sizeof(int);
    float*    tsc    = (float*)(ws + off);    off += (size_t)YB * YTOPK * sizeof(float);
    int*      tidx   = (int*)(ws + off);      off += (size_t)YB * YTOPK * sizeof(int);
    unsigned* maskG  = (unsigned*)(ws + off); off += (size_t)YB * YTOPK * 32 * sizeof(unsigned);
    float*    dets   = (float*)(ws + off);    off += (size_t)YB * YMAXDET * 6 * sizeof(float);
    float*    bstats = (float*)(ws + off);    off += (size_t)YB * 2 * sizeof(float);

    const int ntiles = (YB * YN + YTILE - 1) / YTILE;
    score_kernel<<<ntiles, 256, 0, stream>>>(in, scores, cls);
    topk_kernel<<<YB, 256, 0, stream>>>(scores, tsc, tidx);
    nms_kernel<<<YB, 1024, 0, stream>>>(in, tsc, tidx, cls, maskG, dets, bstats);
    finalize_kernel<<<1, 256, 0, stream>>>(dets, bstats, out);
}